// QAOAInitialiserGNN_41875931136205
// MI455X (gfx1250) — compile-verified
//
#include <hip/hip_runtime.h>
#include <hip/hip_bf16.h>

typedef float v2f __attribute__((ext_vector_type(2)));
typedef float v8f __attribute__((ext_vector_type(8)));

#define PI_F 3.14159265358979323846f

__device__ __forceinline__ float sigmoidf(float x) { return 1.0f / (1.0f + __expf(-x)); }

// ---------------------------------------------------------------------------
// small utility kernels
// ---------------------------------------------------------------------------
__global__ void k_fill(float* __restrict__ p, float v, int n) {
    int i = blockIdx.x * blockDim.x + threadIdx.x;
    if (i < n) p[i] = v;
}

// degree accumulation over edge destinations (deg pre-initialized to 1.0 = self loop)
__global__ void k_deg(const int* __restrict__ dst, float* __restrict__ deg, int nE) {
    int e = blockIdx.x * blockDim.x + threadIdx.x;
    if (e < nE) atomicAdd(&deg[dst[e]], 1.0f);
}

__global__ void k_rsqrt_inplace(float* __restrict__ p, int n) {
    int i = blockIdx.x * blockDim.x + threadIdx.x;
    if (i < n) p[i] = rsqrtf(p[i]);  // deg >= 1 always (self loop), matches reference
}

// conv1 feature transform: h[n][j] = x[n] * W1[j]   (x is [N,1], W1 is [1,32])
__global__ void k_layer1(const float* __restrict__ x, const float* __restrict__ W1,
                         float* __restrict__ h, int nN) {
    int i = blockIdx.x * blockDim.x + threadIdx.x;
    int n = i >> 5, j = i & 31;
    if (n < nN) h[i] = x[n] * W1[j];
}

// init conv accumulator with self-loop message + bias: out[n][j] = hW[n][j]*dinv[n]^2 + b[j]
__global__ void k_selfloop(const float* __restrict__ hW, const float* __restrict__ dinv,
                           const float* __restrict__ b, float* __restrict__ out, int nN) {
    int i = blockIdx.x * blockDim.x + threadIdx.x;
    int n = i >> 5, j = i & 31;
    if (n < nN) {
        float di = dinv[n];
        out[i] = hW[i] * di * di + b[j];
    }
}

// edge message scatter: one wave32 per edge, lane j = feature j (coalesced 128B rows)
__global__ void k_scatter(const float* __restrict__ hW, const int* __restrict__ src,
                          const int* __restrict__ dst, const float* __restrict__ dinv,
                          float* __restrict__ out, int nE) {
    int e = (blockIdx.x * blockDim.x + threadIdx.x) >> 5;
    int lane = threadIdx.x & 31;
    if (e >= nE) return;
    int s = src[e], d = dst[e];
    float w = dinv[s] * dinv[d];
    atomicAdd(&out[d * 32 + lane], hW[s * 32 + lane] * w);
}

// ---------------------------------------------------------------------------
// WMMA fp32 matmul: Y[r,0:32] = (relu_in ? relu(X) : X)[r,0:32] @ W[32,32] (+bias)(relu_out)
// One wave per 16 output rows, two 16x16 N-tiles, 8 K-chunks of V_WMMA_F32_16X16X4_F32.
// A layout (16x4 f32): lanes 0-15 rows M=0..15 hold K={0,1}; lanes 16-31 hold K={2,3}.
// B layout (4x16 f32): VGPR0 lanes 0-15 = K0, lanes 16-31 = K2; VGPR1 = K1/K3.
// C/D layout (16x16 f32): elem v, lanes 0-15 -> row v, lanes 16-31 -> row v+8; col = lane%16.
// ---------------------------------------------------------------------------
__global__ void k_mm32_wmma(const float* __restrict__ X, const float* __restrict__ W,
                            const float* __restrict__ bias, float* __restrict__ Y,
                            int nrows, int relu_in, int relu_out) {
    int wave = (blockIdx.x * blockDim.x + threadIdx.x) >> 5;
    int lane = threadIdx.x & 31;
    int row0 = wave * 16;
    if (row0 >= nrows) return;   // wave-uniform exit: EXEC all-ones inside WMMA
    int half = lane >> 4;
    int l16  = lane & 15;

    int arow = row0 + l16;
    if (arow >= nrows) arow = nrows - 1;  // clamp (values masked by store guard)
    const float* xr = X + (size_t)arow * 32;

    v8f c0 = {0.f, 0.f, 0.f, 0.f, 0.f, 0.f, 0.f, 0.f};
    v8f c1 = {0.f, 0.f, 0.f, 0.f, 0.f, 0.f, 0.f, 0.f};

#pragma unroll
    for (int kk = 0; kk < 8; ++kk) {
        int k0 = kk * 4 + half * 2;
        v2f a;
        a.x = xr[k0];
        a.y = xr[k0 + 1];
        if (relu_in) { a.x = fmaxf(a.x, 0.f); a.y = fmaxf(a.y, 0.f); }
        v2f b0, b1;
        b0.x = W[k0 * 32 + l16];       b0.y = W[(k0 + 1) * 32 + l16];
        b1.x = W[k0 * 32 + 16 + l16];  b1.y = W[(k0 + 1) * 32 + 16 + l16];
        // (neg_a, A, neg_b, B, c_mod, C, reuse_a, reuse_b)
        c0 = __builtin_amdgcn_wmma_f32_16x16x4_f32(false, a, false, b0, (short)0, c0, false, false);
        c1 = __builtin_amdgcn_wmma_f32_16x16x4_f32(false, a, false, b1, (short)0, c1, false, false);
    }

    float blo = bias ? bias[l16] : 0.f;
    float bhi = bias ? bias[16 + l16] : 0.f;
#pragma unroll
    for (int v = 0; v < 8; ++v) {
        int r = row0 + v + half * 8;
        if (r < nrows) {
            float y0 = c0[v] + blo;
            float y1 = c1[v] + bhi;
            if (relu_out) { y0 = fmaxf(y0, 0.f); y1 = fmaxf(y1, 0.f); }
            Y[(size_t)r * 32 + l16]      = y0;
            Y[(size_t)r * 32 + 16 + l16] = y1;
        }
    }
}

// theta[n] = PI * sigmoid(dot(z1[n,:], Wt2) + bt2)   (one wave per node)
__global__ void k_theta(const float* __restrict__ z1, const float* __restrict__ Wt2,
                        const float* __restrict__ bt2, float* __restrict__ theta, int nN) {
    int n = (blockIdx.x * blockDim.x + threadIdx.x) >> 5;
    int lane = threadIdx.x & 31;
    if (n >= nN) return;
    float v = z1[n * 32 + lane] * Wt2[lane];
#pragma unroll
    for (int o = 16; o > 0; o >>= 1) v += __shfl_down(v, o, 32);
    if (lane == 0) theta[n] = PI_F * sigmoidf(v + bt2[0]);
}

// per-graph pooling of relu(h): one wave per node
__global__ void k_pool(const float* __restrict__ h, const int* __restrict__ batch,
                       float* __restrict__ gsum, float* __restrict__ counts, int nN) {
    int n = (blockIdx.x * blockDim.x + threadIdx.x) >> 5;
    int lane = threadIdx.x & 31;
    if (n >= nN) return;
    int g = batch[n];
    atomicAdd(&gsum[g * 32 + lane], fmaxf(h[n * 32 + lane], 0.f));
    if (lane == 0) atomicAdd(&counts[g], 1.0f);
}

__global__ void k_gemb(const float* __restrict__ gsum, const float* __restrict__ counts,
                       float* __restrict__ gemb, int nG) {
    int i = blockIdx.x * blockDim.x + threadIdx.x;
    if (i < nG * 32) gemb[i] = gsum[i] / fmaxf(counts[i >> 5], 1.0f);
}

// beta_gamma[g,:] = 2*PI*sigmoid(zg[g,:] @ Wg2[32,2] + bg2)   (one wave per graph)
__global__ void k_betagamma(const float* __restrict__ zg, const float* __restrict__ Wg2,
                            const float* __restrict__ bg2, float* __restrict__ outBG, int nG) {
    int g = (blockIdx.x * blockDim.x + threadIdx.x) >> 5;
    int lane = threadIdx.x & 31;
    if (g >= nG) return;
    float v  = zg[g * 32 + lane];
    float d0 = v * Wg2[lane * 2 + 0];
    float d1 = v * Wg2[lane * 2 + 1];
#pragma unroll
    for (int o = 16; o > 0; o >>= 1) {
        d0 += __shfl_down(d0, o, 32);
        d1 += __shfl_down(d1, o, 32);
    }
    if (lane == 0) {
        outBG[g * 2 + 0] = 2.0f * PI_F * sigmoidf(d0 + bg2[0]);
        outBG[g * 2 + 1] = 2.0f * PI_F * sigmoidf(d1 + bg2[1]);
    }
}

// ---------------------------------------------------------------------------
extern "C" void kernel_launch(void* const* d_in, const int* in_sizes, int n_in,
                              void* d_out, int out_size, void* d_ws, size_t ws_size,
                              hipStream_t stream) {
    (void)n_in; (void)ws_size;
    const float* x     = (const float*)d_in[0];
    const int*   ei    = (const int*)d_in[1];
    const int*   batch = (const int*)d_in[2];
    const float* W1    = (const float*)d_in[3];
    const float* b1    = (const float*)d_in[4];
    const float* W2    = (const float*)d_in[5];
    const float* b2    = (const float*)d_in[6];
    const float* W3    = (const float*)d_in[7];
    const float* b3    = (const float*)d_in[8];
    const float* Wt1   = (const float*)d_in[9];
    const float* bt1   = (const float*)d_in[10];
    const float* Wt2   = (const float*)d_in[11];
    const float* bt2   = (const float*)d_in[12];
    const float* Wg1   = (const float*)d_in[13];
    const float* bg1   = (const float*)d_in[14];
    const float* Wg2   = (const float*)d_in[15];
    const float* bg2   = (const float*)d_in[16];

    const int N = in_sizes[0];
    const int E = in_sizes[1] / 2;
    const int G = (out_size - N) / 2;

    const int* src = ei;
    const int* dst = ei + E;

    // workspace carve-up (floats)
    float* ws     = (float*)d_ws;
    float* dinv   = ws;  ws += N;               // deg, then rsqrt in place
    float* bufA   = ws;  ws += (size_t)N * 32;  // transformed features h@W (message source)
    float* bufB   = ws;  ws += (size_t)N * 32;  // conv accumulator / layer output
    float* gsum   = ws;  ws += (size_t)G * 32;  // pooled sums  (contiguous with counts!)
    float* counts = ws;  ws += G;
    float* gemb   = ws;  ws += (size_t)G * 32;
    float* zg     = ws;  ws += (size_t)G * 32;

    float* theta = (float*)d_out;
    float* outBG = (float*)d_out + N;

    const int T = 256;
    auto cdiv = [](long long a, long long b) { return (int)((a + b - 1) / b); };

    const int nodeElemBlk = cdiv((long long)N * 32, T);              // thread per (n,j) / wave per node
    const int edgeWaveBlk = cdiv((long long)E * 32, T);              // wave per edge
    const int mmBlkN      = cdiv(((long long)N + 15) / 16 * 32, T);  // wave per 16 rows
    const int mmBlkG      = cdiv(((long long)G + 15) / 16 * 32, T);

    // --- GCN normalization: dinv = rsqrt(1 + in-degree) ---
    k_fill<<<cdiv(N, T), T, 0, stream>>>(dinv, 1.0f, N);
    k_deg<<<cdiv(E, T), T, 0, stream>>>(dst, dinv, E);
    k_rsqrt_inplace<<<cdiv(N, T), T, 0, stream>>>(dinv, N);

    // --- conv1: h = x@W1; out = segsum(h[src]*norm) + b1 ---
    k_layer1<<<nodeElemBlk, T, 0, stream>>>(x, W1, bufA, N);
    k_selfloop<<<nodeElemBlk, T, 0, stream>>>(bufA, dinv, b1, bufB, N);
    k_scatter<<<edgeWaveBlk, T, 0, stream>>>(bufA, src, dst, dinv, bufB, E);

    // --- conv2: relu fused into WMMA input read ---
    k_mm32_wmma<<<mmBlkN, T, 0, stream>>>(bufB, W2, nullptr, bufA, N, 1, 0);
    k_selfloop<<<nodeElemBlk, T, 0, stream>>>(bufA, dinv, b2, bufB, N);
    k_scatter<<<edgeWaveBlk, T, 0, stream>>>(bufA, src, dst, dinv, bufB, E);

    // --- conv3 ---
    k_mm32_wmma<<<mmBlkN, T, 0, stream>>>(bufB, W3, nullptr, bufA, N, 1, 0);
    k_selfloop<<<nodeElemBlk, T, 0, stream>>>(bufA, dinv, b3, bufB, N);
    k_scatter<<<edgeWaveBlk, T, 0, stream>>>(bufA, src, dst, dinv, bufB, E);
    // h := relu(bufB), applied at every read site below

    // --- theta head: z1 = relu(relu(h)@Wt1 + bt1); theta = PI*sigmoid(z1@Wt2 + bt2) ---
    k_mm32_wmma<<<mmBlkN, T, 0, stream>>>(bufB, Wt1, bt1, bufA, N, 1, 1);
    k_theta<<<nodeElemBlk, T, 0, stream>>>(bufA, Wt2, bt2, theta, N);

    // --- global mean pool + graph head ---
    k_fill<<<cdiv((long long)G * 33, T), T, 0, stream>>>(gsum, 0.0f, G * 33);  // gsum + counts
    k_pool<<<nodeElemBlk, T, 0, stream>>>(bufB, batch, gsum, counts, N);
    k_gemb<<<cdiv((long long)G * 32, T), T, 0, stream>>>(gsum, counts, gemb, G);
    k_mm32_wmma<<<mmBlkG, T, 0, stream>>>(gemb, Wg1, bg1, zg, G, 0, 1);
    k_betagamma<<<cdiv((long long)G * 32, T), T, 0, stream>>>(zg, Wg2, bg2, outBG, G);
}